// EIGWhiteningSigma_19963007992128
// MI455X (gfx1250) — compile-verified
//
#include <hip/hip_runtime.h>

// ---------------- problem constants ----------------
#define GROUPS 16
#define D      16            // channels per whitening group
#define BATCH  16
#define CHN    256
#define LEN    16384
#define MTOT   (BATCH * LEN) // 262144 samples per group

#define TILE_COLS        128
#define TILES_PER_BLOCK  16
#define COLS_PER_BLOCK   (TILE_COLS * TILES_PER_BLOCK)   // 2048
#define CHUNKS_PER_GROUP 128                              // 16 batches * 8 L-chunks
#define LSTRIDE          132                              // padded LDS row stride (bank-conflict free)
#define EPSV             1e-3f
#define NS_ITERS         15

typedef float v2f __attribute__((ext_vector_type(2)));
typedef float v8f __attribute__((ext_vector_type(8)));

// ---------------- workspace layout (floats) ----------------
#define WS_COV   0                   // [16][256] raw second moments  (zeroed each launch)
#define WS_SUM   (16 * 256)          // [16][16]  per-channel sums    (zeroed each launch)
#define WS_WM    (WS_SUM + 16 * 16)  // [16][256] whitening matrices
#define WS_MEAN  (WS_WM + 16 * 256)  // [16][16]  means
#define WS_ZERO_COUNT (WS_WM)        // floats to zero per launch

__device__ __forceinline__ v8f wmma4(v2f a, v2f b, v8f c) {
  // D(16x16,f32) += A(16x4,f32) * B(4x16,f32)
  return __builtin_amdgcn_wmma_f32_16x16x4_f32(false, a, false, b, (short)0, c,
                                               false, false);
}

// ---------------- kernel 0: zero accumulators ----------------
__global__ void __launch_bounds__(256) zca_zero(float* __restrict__ ws) {
  const int t = blockIdx.x * 256 + threadIdx.x;
  if (t < WS_ZERO_COUNT) ws[t] = 0.0f;
}

// ---------------- kernel 1: raw covariance + sums via WMMA ----------------
__global__ void __launch_bounds__(256) zca_stats(const float* __restrict__ X,
                                                 float* __restrict__ ws) {
  __shared__ __attribute__((aligned(16))) float tile[D * LSTRIDE];
  __shared__ float redCov[8 * 256];
  __shared__ float redSum[256];

  const int g     = blockIdx.y;
  const int chunk = blockIdx.x;            // 0..127
  const int b     = chunk >> 3;            // batch index
  const int l0b   = (chunk & 7) * COLS_PER_BLOCK;
  const int t     = threadIdx.x;
  const int lane  = t & 31;
  const int w     = t >> 5;                // wave id, 0..7
  const int hi    = lane >> 4;             // lane half
  const int nlo   = lane & 15;

  const size_t rowBase = ((size_t)b * CHN + (size_t)g * D) * (size_t)LEN;

  v8f   acc  = {};
  float csum = 0.0f;

  for (int ti = 0; ti < TILES_PER_BLOCK; ++ti) {
    const int l0 = l0b + ti * TILE_COLS;
    // ---- stage 16 x 128 tile, fully coalesced float4 loads ----
    {
      int q = t, r = q >> 5, c4 = q & 31;
      *(float4*)&tile[r * LSTRIDE + c4 * 4] =
          *(const float4*)(X + rowBase + (size_t)r * LEN + l0 + c4 * 4);
      q = t + 256; r = q >> 5; c4 = q & 31;
      *(float4*)&tile[r * LSTRIDE + c4 * 4] =
          *(const float4*)(X + rowBase + (size_t)r * LEN + l0 + c4 * 4);
    }
    __syncthreads();

    // ---- per-channel partial sums (thread t: channel t>>4, 8 cols) ----
    {
      const int r = t >> 4, c0 = (t & 15) * 8;
#pragma unroll
      for (int j = 0; j < 8; ++j) csum += tile[r * LSTRIDE + c0 + j];
    }

    // ---- wave w accumulates 16 K-columns of the outer product ----
    // A-layout frag (M=nlo, K = k0 + 2*hi + vgpr) equals the B-layout frag
    // of the transposed operand -> same registers feed both A and B.
    const int colBase = w * 16;
#pragma unroll
    for (int k0 = 0; k0 < 16; k0 += 4) {
      const int col = colBase + k0 + 2 * hi;
      const v2f f   = *(const v2f*)&tile[nlo * LSTRIDE + col];
      acc = wmma4(f, f, acc);
    }
    __syncthreads();
  }

  // ---- reduce 8 wave accumulators, then atomic into ws ----
#pragma unroll
  for (int r = 0; r < 8; ++r) {
    const int e = (r + 8 * hi) * 16 + nlo;   // C-layout -> row-major element
    redCov[w * 256 + e] = acc[r];
  }
  redSum[t] = csum;
  __syncthreads();

  float s = 0.0f;
#pragma unroll
  for (int wv = 0; wv < 8; ++wv) s += redCov[wv * 256 + t];
  atomicAdd(&ws[WS_COV + g * 256 + t], s);

  if (t < D) {
    float cs = 0.0f;
#pragma unroll
    for (int j = 0; j < 16; ++j) cs += redSum[t * 16 + j];
    atomicAdd(&ws[WS_SUM + g * D + t], cs);
  }
}

// ---------------- kernel 2: sigma^(-1/2) via Newton-Schulz ----------------
__global__ void __launch_bounds__(256) zca_solve(float* __restrict__ ws) {
  __shared__ float Ym[256], Zm[256], Tm[256];
  __shared__ float trS;

  const int g = blockIdx.x;
  const int t = threadIdx.x;
  const int i = t >> 4, j = t & 15;
  const float invm = 1.0f / (float)MTOT;

  const float mi  = ws[WS_SUM + g * D + i] * invm;
  const float mj  = ws[WS_SUM + g * D + j] * invm;
  const float sig = ws[WS_COV + g * 256 + t] * invm - mi * mj +
                    ((i == j) ? EPSV : 0.0f);
  Tm[t] = sig;
  __syncthreads();
  if (t == 0) {
    float tr = 0.0f;
    for (int k = 0; k < D; ++k) tr += Tm[k * 17];
    trS = tr;
  }
  __syncthreads();
  const float tr = trS;
  Ym[t] = sig / tr;                    // Y0 = sigma / tr  (spec in (0,1])
  Zm[t] = (i == j) ? 1.0f : 0.0f;      // Z0 = I
  __syncthreads();

  for (int it = 0; it < NS_ITERS; ++it) {
    float zy = 0.0f;
#pragma unroll
    for (int k = 0; k < D; ++k) zy += Zm[i * 16 + k] * Ym[k * 16 + j];
    Tm[t] = ((i == j) ? 1.5f : 0.0f) - 0.5f * zy;   // T = (3I - ZY)/2
    __syncthreads();
    float ny = 0.0f, nz = 0.0f;
#pragma unroll
    for (int k = 0; k < D; ++k) {
      ny += Ym[i * 16 + k] * Tm[k * 16 + j];        // Y <- Y T
      nz += Tm[i * 16 + k] * Zm[k * 16 + j];        // Z <- T Z
    }
    __syncthreads();
    Ym[t] = ny;
    Zm[t] = nz;
    __syncthreads();
  }

  ws[WS_WM + g * 256 + t] = Zm[t] * rsqrtf(tr);     // sigma^{-1/2}
  if (j == 0) ws[WS_MEAN + g * D + i] = mi;
}

// ---------------- kernel 3: apply wm * (x - mean) via WMMA ----------------
__global__ void __launch_bounds__(256) zca_apply(const float* __restrict__ X,
                                                 const float* __restrict__ weight,
                                                 const float* __restrict__ bias,
                                                 const float* __restrict__ ws,
                                                 float* __restrict__ out) {
  __shared__ __attribute__((aligned(16))) float tile[D * LSTRIDE];
  __shared__ float wmS[256];
  __shared__ float meanS[16], wS[16], bS[16];

  const int g     = blockIdx.y;
  const int chunk = blockIdx.x;
  const int b     = chunk >> 3;
  const int l0b   = (chunk & 7) * COLS_PER_BLOCK;
  const int t     = threadIdx.x;
  const int lane  = t & 31;
  const int w     = t >> 5;
  const int hi    = lane >> 4;
  const int nlo   = lane & 15;

  wmS[t] = ws[WS_WM + g * 256 + t];
  if (t < D) {
    meanS[t] = ws[WS_MEAN + g * D + t];
    wS[t]    = weight[g * D + t];
    bS[t]    = bias[g * D + t];
  }
  __syncthreads();

  // preload whitening matrix as 4 A-fragments (A[m][k] = wm[m][k0+k])
  v2f afrag[4];
#pragma unroll
  for (int kk = 0; kk < 4; ++kk)
    afrag[kk] = *(const v2f*)&wmS[nlo * 16 + kk * 4 + 2 * hi];

  const size_t rowBase = ((size_t)b * CHN + (size_t)g * D) * (size_t)LEN;

  for (int ti = 0; ti < TILES_PER_BLOCK; ++ti) {
    const int l0 = l0b + ti * TILE_COLS;
    {
      int q = t, r = q >> 5, c4 = q & 31;
      *(float4*)&tile[r * LSTRIDE + c4 * 4] =
          *(const float4*)(X + rowBase + (size_t)r * LEN + l0 + c4 * 4);
      q = t + 256; r = q >> 5; c4 = q & 31;
      *(float4*)&tile[r * LSTRIDE + c4 * 4] =
          *(const float4*)(X + rowBase + (size_t)r * LEN + l0 + c4 * 4);
    }
    __syncthreads();

    const int colBase = w * 16;   // wave's 16 output columns
    v8f acc = {};
#pragma unroll
    for (int kk = 0; kk < 4; ++kk) {
      const int kr = kk * 4 + 2 * hi;   // B-layout rows for this VGPR pair
      v2f bf;
      bf.x = tile[kr * LSTRIDE + colBase + nlo] - meanS[kr];
      bf.y = tile[(kr + 1) * LSTRIDE + colBase + nlo] - meanS[kr + 1];
      acc = wmma4(afrag[kk], bf, acc);
    }

    // D is in C-layout: row = r + 8*hi, col = nlo
    float* outBase = out + rowBase + (size_t)l0 + colBase + nlo;
#pragma unroll
    for (int r = 0; r < 8; ++r) {
      const int ch = r + 8 * hi;
      outBase[(size_t)ch * LEN] = acc[r] * wS[ch] + bS[ch];
    }
    __syncthreads();
  }
}

// ---------------- launch ----------------
extern "C" void kernel_launch(void* const* d_in, const int* in_sizes, int n_in,
                              void* d_out, int out_size, void* d_ws,
                              size_t ws_size, hipStream_t stream) {
  const float* X      = (const float*)d_in[0];
  const float* weight = (const float*)d_in[1];
  const float* bias   = (const float*)d_in[2];
  float* out = (float*)d_out;
  float* ws  = (float*)d_ws;

  zca_zero<<<(WS_ZERO_COUNT + 255) / 256, 256, 0, stream>>>(ws);
  zca_stats<<<dim3(CHUNKS_PER_GROUP, GROUPS), 256, 0, stream>>>(X, ws);
  zca_solve<<<GROUPS, 256, 0, stream>>>(ws);
  zca_apply<<<dim3(CHUNKS_PER_GROUP, GROUPS), 256, 0, stream>>>(X, weight, bias,
                                                                ws, out);
}